// CenterLoss_90245852823755
// MI455X (gfx1250) — compile-verified
//
#include <hip/hip_runtime.h>

#define NUM_CLASSES 10000
#define FEATURE_DIM 2048
#define BATCH       512
#define ALPHA       0.5f

typedef float v2f __attribute__((ext_vector_type(2)));
typedef float v4f __attribute__((ext_vector_type(4)));
typedef float v8f __attribute__((ext_vector_type(8)));
// 4-byte-aligned float4: legal for B128 access in DWORD alignment mode,
// used for the phase-shifted (d_out+1) destination stream.
typedef float v4f_u __attribute__((ext_vector_type(4), aligned(4)));

// ---------------------------------------------------------------------------
// Kernel 1: stream-copy centers -> out_centers. Loads are 16B-aligned B128;
// stores are dword-aligned B128 (destination is offset by one float). Non-
// temporal hints keep the 160MB stream from displacing lines the update/loss
// kernels still need.
// ---------------------------------------------------------------------------
__global__ void cl_copy_centers(const v4f* __restrict__ src,
                                v4f_u* __restrict__ dst, long n4) {
    long i = (long)blockIdx.x * blockDim.x + threadIdx.x;
    if (i < n4) {
        v4f v = __builtin_nontemporal_load(src + i);
        __builtin_nontemporal_store((v4f_u)v, dst + i);
    }
}

// ---------------------------------------------------------------------------
// Kernel 2: sequential EMA update. Block b handles label labels[b] only if b
// is the FIRST occurrence of that label; it then applies the whole chain of
// updates for that label in batch order (exact duplicate-compounding
// semantics). Reads ORIGINAL centers, overwrites the copied row.
// 512 threads * 4 floats = 2048 columns.
// ---------------------------------------------------------------------------
__global__ void cl_update_centers(const float* __restrict__ features,
                                  const int* __restrict__ labels,
                                  const float* __restrict__ centers,
                                  float* __restrict__ out_centers) {
    __shared__ int slab[BATCH];
    for (int i = threadIdx.x; i < BATCH; i += blockDim.x) slab[i] = labels[i];
    __syncthreads();

    const int b = blockIdx.x;
    const int l = slab[b];
    for (int j = 0; j < b; ++j)
        if (slab[j] == l) return;          // uniform across block: safe exit

    const int  col  = threadIdx.x * 4;
    const long coff = (long)l * FEATURE_DIM + col;
    v4f c = *(const v4f*)(centers + coff);  // 16B-aligned

    for (int j = b; j < BATCH; ++j) {
        if (slab[j] == l) {
            v4f f = *(const v4f*)(features + (long)j * FEATURE_DIM + col);
            c = c + ALPHA * (f - c);
        }
    }
    // out row is phase-shifted by 1 float -> dword-aligned B128 store
    *(v4f_u*)(out_centers + coff) = (v4f_u)c;
}

// ---------------------------------------------------------------------------
// Kernel 3: per-row loss partial sums using V_WMMA_F32_16X16X4_F32.
// A-tile holds diff^2 (64 f32 per wave per issue), B = all-ones, so
// D[m][n] = rowsum(A) for every n; total over all D slots = 16 * sum(A).
// Accumulator carries across the 4 issues per wave; fixed-order LDS tree
// reduction afterwards keeps the result deterministic.
// ---------------------------------------------------------------------------
__global__ void cl_loss_partial(const float* __restrict__ features,
                                const int* __restrict__ labels,
                                const float* __restrict__ centers,
                                float* __restrict__ partial) {
    const int  b    = blockIdx.x;                 // batch row
    const int  l    = labels[b];
    const float* frow = features + (long)b * FEATURE_DIM;
    const float* crow = centers  + (long)l * FEATURE_DIM;

    const int wave = threadIdx.x >> 5;            // 8 waves of 32
    const int lane = threadIdx.x & 31;
    const int wbase = wave * (FEATURE_DIM / 8);   // 256 elements per wave

    v8f acc = {};
    v2f ones; ones.x = 1.0f; ones.y = 1.0f;

#pragma unroll
    for (int t = 0; t < 4; ++t) {                 // 4 * 64 = 256 elements
        int idx = wbase + t * 64 + lane * 2;      // 8B-aligned
        v2f f = *(const v2f*)(frow + idx);
        v2f c = *(const v2f*)(crow + idx);
        v2f d; d.x = f.x - c.x; d.y = f.y - c.y;
        v2f a; a.x = d.x * d.x; a.y = d.y * d.y;
        acc = __builtin_amdgcn_wmma_f32_16x16x4_f32(
            /*neg_a=*/false, a, /*neg_b=*/false, ones,
            /*c_mod=*/(short)0, acc, /*reuse_a=*/false, /*reuse_b=*/false);
    }

    float s = acc[0] + acc[1] + acc[2] + acc[3] +
              acc[4] + acc[5] + acc[6] + acc[7];

    __shared__ float red[256];
    red[threadIdx.x] = s;
    __syncthreads();
    for (int off = 128; off > 0; off >>= 1) {
        if ((int)threadIdx.x < off) red[threadIdx.x] += red[threadIdx.x + off];
        __syncthreads();
    }
    if (threadIdx.x == 0) partial[b] = red[0] * (1.0f / 16.0f);
}

// ---------------------------------------------------------------------------
// Kernel 4: deterministic final reduction of 512 partials -> mean.
// ---------------------------------------------------------------------------
__global__ void cl_loss_final(const float* __restrict__ partial,
                              float* __restrict__ out) {
    __shared__ float red[256];
    const int t = threadIdx.x;
    red[t] = partial[t] + partial[t + 256];
    __syncthreads();
    for (int off = 128; off > 0; off >>= 1) {
        if (t < off) red[t] += red[t + off];
        __syncthreads();
    }
    if (t == 0) out[0] = red[0] * (1.0f / (float)(BATCH * FEATURE_DIM));
}

extern "C" void kernel_launch(void* const* d_in, const int* in_sizes, int n_in,
                              void* d_out, int out_size, void* d_ws, size_t ws_size,
                              hipStream_t stream) {
    const float* features = (const float*)d_in[0];   // (512, 2048) f32
    const int*   labels   = (const int*)d_in[1];     // (512,) int
    const float* centers  = (const float*)d_in[2];   // (10000, 2048) f32

    float* out         = (float*)d_out;      // out[0] = loss
    float* out_centers = out + 1;            // out[1:] = new_centers
    float* partial     = (float*)d_ws;       // 512 floats of scratch

    const long n  = (long)NUM_CLASSES * FEATURE_DIM;   // 20,480,000
    const long n4 = n / 4;                             // 5,120,000 (exact)

    // 1) 80MB streaming copy centers -> out_centers (B128 both sides)
    cl_copy_centers<<<(int)((n4 + 255) / 256), 256, 0, stream>>>(
        (const v4f*)centers, (v4f_u*)out_centers, n4);

    // 2) sequential EMA chains (overwrites copied rows; stream-ordered)
    cl_update_centers<<<BATCH, FEATURE_DIM / 4, 0, stream>>>(
        features, labels, centers, out_centers);

    // 3) WMMA-based per-row squared-error partials
    cl_loss_partial<<<BATCH, 256, 0, stream>>>(features, labels, centers, partial);

    // 4) deterministic final mean
    cl_loss_final<<<1, 256, 0, stream>>>(partial, out);
}